// GCRN_57483842289901
// MI455X (gfx1250) — compile-verified
//
#include <hip/hip_runtime.h>
#include <hip/hip_bf16.h>

#define TT 12
#define BB 8
#define NNODE 4096
#define HH 16
#define BN (BB * NNODE)  // 32768 LSTM batch

typedef float v2f __attribute__((ext_vector_type(2)));
typedef float v8f __attribute__((ext_vector_type(8)));

__device__ __forceinline__ float sigf(float v) {
    return 1.0f / (1.0f + __expf(-v));
}
__device__ __forceinline__ float tanhfast(float v) {
    // tanh(x) = 1 - 2/(e^{2x}+1); saturates correctly at +/-inf
    float e = __expf(2.0f * v);
    return 1.0f - 2.0f / (e + 1.0f);
}

// 16-wide dot: w points at LDS row (64B aligned), v is a register array
__device__ __forceinline__ float dot16(const float* __restrict__ w,
                                       const float* __restrict__ v) {
    float acc = 0.0f;
#pragma unroll
    for (int q = 0; q < 4; ++q) {
        float4 ww = *(const float4*)(w + 4 * q);
        acc += ww.x * v[4 * q + 0] + ww.y * v[4 * q + 1] +
               ww.z * v[4 * q + 2] + ww.w * v[4 * q + 3];
    }
    return acc;
}

// ---------------------------------------------------------------------------
// Kernel 1: fused sigmoid(x) -> 12-step LSTM -> temp(h), res_D, P = Wg h + bg
// One thread per sequence; weights staged in LDS (broadcast b128 reads).
// ---------------------------------------------------------------------------
__global__ __launch_bounds__(256) void lstm_head_kernel(
    const float* __restrict__ x,      // (T, BN, 16)
    const float* __restrict__ W_ih,   // (64,16)
    const float* __restrict__ W_hh,   // (64,16)
    const float* __restrict__ b_ih,   // (64)
    const float* __restrict__ b_hh,   // (64)
    const float* __restrict__ Wd,     // (1,16)
    const float* __restrict__ bd,     // (1)
    const float* __restrict__ Wg,     // (16,16)
    const float* __restrict__ bg,     // (16)
    float* __restrict__ temp_out,     // (BN,16)
    float* __restrict__ P_out,        // (BN,16)
    float* __restrict__ resD)         // (BN)
{
    __shared__ __align__(16) float sWih[64][16];
    __shared__ __align__(16) float sWhh[64][16];
    __shared__ __align__(16) float sWg[16][16];
    __shared__ float sBias[64];
    __shared__ float sWd[16];
    __shared__ float sBg[16];

    const int tid = threadIdx.x;
    for (int idx = tid; idx < 1024; idx += 256) {
        sWih[idx >> 4][idx & 15] = W_ih[idx];
        sWhh[idx >> 4][idx & 15] = W_hh[idx];
    }
    if (tid < 64) sBias[tid] = b_ih[tid] + b_hh[tid];
    if (tid < 256) {
        int idx = tid;
        if (idx < 256) sWg[idx >> 4][idx & 15] = Wg[idx];
    }
    if (tid < 16) { sWd[tid] = Wd[tid]; sBg[tid] = bg[tid]; }
    __syncthreads();

    const int s = blockIdx.x * 256 + tid;  // 0..32767

    float h[16], c[16], hn[16], xv[16];
#pragma unroll
    for (int k = 0; k < 16; ++k) { h[k] = 0.0f; c[k] = 0.0f; }

    for (int t = 0; t < TT; ++t) {
        const float4* xp = (const float4*)(x + ((size_t)t * BN + s) * HH);
#pragma unroll
        for (int q = 0; q < 4; ++q) {
            float4 v = xp[q];
            xv[4 * q + 0] = sigf(v.x);
            xv[4 * q + 1] = sigf(v.y);
            xv[4 * q + 2] = sigf(v.z);
            xv[4 * q + 3] = sigf(v.w);
        }
#pragma unroll
        for (int j = 0; j < 16; ++j) {
            float gi = sBias[j]      + dot16(&sWih[j][0], xv)      + dot16(&sWhh[j][0], h);
            float gf = sBias[16 + j] + dot16(&sWih[16 + j][0], xv) + dot16(&sWhh[16 + j][0], h);
            float gg = sBias[32 + j] + dot16(&sWih[32 + j][0], xv) + dot16(&sWhh[32 + j][0], h);
            float go = sBias[48 + j] + dot16(&sWih[48 + j][0], xv) + dot16(&sWhh[48 + j][0], h);
            float cj = sigf(gf) * c[j] + sigf(gi) * tanhfast(gg);
            c[j] = cj;
            hn[j] = sigf(go) * tanhfast(cj);
        }
#pragma unroll
        for (int k = 0; k < 16; ++k) h[k] = hn[k];
    }

    // res_D = Wd . h + bd
    resD[s] = bd[0] + dot16(sWd, h);

    // temp = h
    float4* to = (float4*)(temp_out + (size_t)s * 16);
#pragma unroll
    for (int q = 0; q < 4; ++q)
        to[q] = make_float4(h[4 * q], h[4 * q + 1], h[4 * q + 2], h[4 * q + 3]);

    // P = Wg h + bg
    float pv[16];
#pragma unroll
    for (int j = 0; j < 16; ++j) pv[j] = sBg[j] + dot16(&sWg[j][0], h);
    float4* po = (float4*)(P_out + (size_t)s * 16);
#pragma unroll
    for (int q = 0; q < 4; ++q)
        po[q] = make_float4(pv[4 * q], pv[4 * q + 1], pv[4 * q + 2], pv[4 * q + 3]);
}

// ---------------------------------------------------------------------------
// Kernel 2: res_G[b] = P_b (4096x16) @ temp_b^T (16x4096) via
// V_WMMA_F32_16X16X4_F32 chained 4x (K=16). One wave per (b, i-tile, j-chunk);
// A tile reused across 64 j-tiles; NT stores for the 536MB output stream.
// ---------------------------------------------------------------------------
__global__ __launch_bounds__(256) void resg_wmma_kernel(
    const float* __restrict__ P,     // (8,4096,16)
    const float* __restrict__ Tm,    // (8,4096,16)
    float* __restrict__ G)           // (8,4096,4096)
{
    const int lane = threadIdx.x & 31;
    const int wid  = (blockIdx.x * 256 + threadIdx.x) >> 5;  // 0..8191
    const int b    = wid >> 10;          // batch 0..7
    const int r    = wid & 1023;
    const int itle = r >> 2;             // i-tile 0..255
    const int jc   = r & 3;              // j-chunk 0..3 (64 tiles each)

    const float* Pb = P  + (size_t)b * NNODE * HH;
    const float* Tb = Tm + (size_t)b * NNODE * HH;
    float*       Gb = G  + (size_t)b * NNODE * NNODE;

    const int rsel = lane & 15;            // row (A) / col (B,D) within tile
    const int csel = (lane >> 4) << 1;     // k-pair select: 0 or 2
    const int rowoff = (lane >> 4) << 3;   // D rows: r or r+8

    // A = P[i*16 + m, k] in 16x4-chunk layout: lane holds (k=4c+csel, 4c+csel+1)
    v2f a[4];
#pragma unroll
    for (int cc = 0; cc < 4; ++cc) {
        float2 t = *(const float2*)(Pb + ((size_t)(itle * 16 + rsel)) * 16 + 4 * cc + csel);
        a[cc] = (v2f){t.x, t.y};
    }

    const int jt0 = jc * 64;
    for (int jt = jt0; jt < jt0 + 64; ++jt) {
        // B[k][n] = temp[jt*16 + n][k] -> identical per-lane load pattern as A
        v2f bm[4];
#pragma unroll
        for (int cc = 0; cc < 4; ++cc) {
            float2 t = *(const float2*)(Tb + ((size_t)(jt * 16 + rsel)) * 16 + 4 * cc + csel);
            bm[cc] = (v2f){t.x, t.y};
        }

        v8f acc = {0.f, 0.f, 0.f, 0.f, 0.f, 0.f, 0.f, 0.f};
#pragma unroll
        for (int cc = 0; cc < 4; ++cc) {
            acc = __builtin_amdgcn_wmma_f32_16x16x4_f32(
                false, a[cc], false, bm[cc], (short)0, acc, false, false);
        }

        // C/D layout: VGPR rr -> row rr + rowoff, col rsel
        float* out = Gb + (size_t)(itle * 16 + rowoff) * NNODE + (size_t)jt * 16 + rsel;
#pragma unroll
        for (int rr = 0; rr < 8; ++rr) {
            __builtin_nontemporal_store(acc[rr], out + (size_t)rr * NNODE);
        }
    }
}

extern "C" void kernel_launch(void* const* d_in, const int* in_sizes, int n_in,
                              void* d_out, int out_size, void* d_ws, size_t ws_size,
                              hipStream_t stream) {
    const float* x    = (const float*)d_in[0];  // spat_embed (12,8,4096,16)
    const float* W_ih = (const float*)d_in[1];
    const float* W_hh = (const float*)d_in[2];
    const float* b_ih = (const float*)d_in[3];
    const float* b_hh = (const float*)d_in[4];
    const float* Wd   = (const float*)d_in[5];
    const float* bd   = (const float*)d_in[6];
    const float* Wg   = (const float*)d_in[7];
    const float* bg   = (const float*)d_in[8];

    float* resD = (float*)d_out;                 // (8,4096) = 32768
    float* G    = (float*)d_out + BN;            // (8,4096,4096)

    float* temp = (float*)d_ws;                  // (32768,16) = 2MB
    float* Pm   = (float*)d_ws + (size_t)BN * HH;// (32768,16) = 2MB

    lstm_head_kernel<<<BN / 256, 256, 0, stream>>>(
        x, W_ih, W_hh, b_ih, b_hh, Wd, bd, Wg, bg, temp, Pm, resD);

    // 8192 waves: 8 batches x 256 i-tiles x 4 j-chunks, 8 waves/block
    resg_wmma_kernel<<<1024, 256, 0, stream>>>(Pm, temp, G);
}